// DeformableAttention_82368882802778
// MI455X (gfx1250) — compile-verified
//
#include <hip/hip_runtime.h>
#include <hip/hip_bf16.h>
#include <math.h>

typedef __attribute__((ext_vector_type(2))) float v2f;
typedef __attribute__((ext_vector_type(8))) float v8f;
typedef int v4i_g __attribute__((vector_size(16)));   // GNU vector: matches builtin param

// ---------------------------------------------------------------------------
// Problem constants
// ---------------------------------------------------------------------------
#define BB     8
#define NN     1024
#define DIN    1024
#define DOUT   256
#define PP     32
#define NPTS   8
#define ROWS   (BB * NN)          // 8192
#define VWSZ   (NN * DOUT)        // 262144 floats per batch

// ---------------------------------------------------------------------------
// CDNA5 async global->LDS copy (ASYNCcnt path), with safe fallback
// ---------------------------------------------------------------------------
#if __has_builtin(__builtin_amdgcn_global_load_async_to_lds_b128)
#define HAVE_ASYNC_LDS 1
#endif

__device__ __forceinline__ void async_copy_b128(const float* g, float* l) {
#ifdef HAVE_ASYNC_LDS
    __builtin_amdgcn_global_load_async_to_lds_b128(
        (__attribute__((address_space(1))) v4i_g*)g,   // C-style cast drops const
        (__attribute__((address_space(3))) v4i_g*)l,
        0, 0);
#else
    *(float4*)l = *(const float4*)g;
#endif
}

__device__ __forceinline__ void wait_async0() {
#if __has_builtin(__builtin_amdgcn_s_wait_asynccnt)
    __builtin_amdgcn_s_wait_asynccnt(0);
#elif defined(HAVE_ASYNC_LDS)
    asm volatile("s_wait_asynccnt 0" ::: "memory");
#endif
}

// ---------------------------------------------------------------------------
// Pack w_att[1024,8] and w_offset[1024,16] into W1[1024,32] (cols 24..31 = 0)
// ---------------------------------------------------------------------------
__global__ void pack_w1(const float* __restrict__ w_att,
                        const float* __restrict__ w_off,
                        float* __restrict__ W1) {
    int i = blockIdx.x * blockDim.x + threadIdx.x;   // 0 .. 1024*32-1
    if (i >= DIN * 32) return;
    int k = i >> 5;
    int c = i & 31;
    float v = 0.0f;
    if (c < 8)       v = w_att[k * NPTS + c];
    else if (c < 24) v = w_off[k * (NPTS * 2) + (c - 8)];
    W1[i] = v;
}

// ---------------------------------------------------------------------------
// FP32 WMMA GEMM: C[M,N] = A[M,K] * B[K,N], row-major.
// Per wave: 16 x (16*NT) tile via V_WMMA_F32_16X16X4_F32.
// B panel staged in LDS with double-buffered async copies (KC=32 chunk),
// shared by all 8 waves of the block; A chunk prefetched into registers.
// A-frag (16x4): lanes 0-15 carry K = k,k+1 ; lanes 16-31 carry K = k+2,k+3
// B-frag (4x16): VGPR0 = rows k / k+2 (lane halves), VGPR1 = rows k+1 / k+3
// D (16x16 f32): VGPR v -> row m0 + v + 8*(lane>=16), col = n0 + t*16 + lane%16
// ---------------------------------------------------------------------------
template <int NT>
__global__ void gemm_f32_wmma(const float* __restrict__ A,
                              const float* __restrict__ B,
                              float* __restrict__ C,
                              int M, int N, int K) {
    constexpr int TILE_N = 16 * NT;
    constexpr int KC     = 32;                 // K-chunk staged in LDS
    constexpr int LDSROW = TILE_N + 4;         // pad: bank-shift, keeps 16B align
    constexpr int NLOADS = (KC * TILE_N) / (256 * 4);  // b128 per thread per chunk

    __shared__ float Bbuf[2][KC][LDSROW];

    const int t    = threadIdx.x;
    const int wave = t >> 5;
    const int lane = t & 31;
    const int half = lane >> 4;          // 0: lanes 0-15, 1: lanes 16-31
    const int l16  = lane & 15;

    const int m0 = (blockIdx.x * 8 + wave) * 16;
    const int n0 = blockIdx.y * TILE_N;

    const float* __restrict__ Arow = A + (size_t)(m0 + l16) * K + 2 * half;

    v8f acc[NT];
#pragma unroll
    for (int tt = 0; tt < NT; ++tt) acc[tt] = (v8f){0, 0, 0, 0, 0, 0, 0, 0};

    auto issue_chunk = [&](int kc, int buf) {
#pragma unroll
        for (int i = 0; i < NLOADS; ++i) {
            const int e   = (i * 256 + t) * 4;       // flat float index in chunk
            const int row = e / TILE_N;
            const int col = e % TILE_N;
            async_copy_b128(B + (size_t)(kc * KC + row) * N + n0 + col,
                            &Bbuf[buf][row][col]);
        }
    };

    const int nchunks = K / KC;
    issue_chunk(0, 0);

    for (int kc = 0; kc < nchunks; ++kc) {
        wait_async0();          // own async writes done
        __syncthreads();        // everyone's async writes visible
        if (kc + 1 < nchunks) issue_chunk(kc + 1, (kc + 1) & 1);

        // prefetch this wave's A chunk: 8 x b64, batched
        v2f afrag[KC / 4];
#pragma unroll
        for (int j = 0; j < KC / 4; ++j)
            afrag[j] = *(const v2f*)(Arow + kc * KC + 4 * j);

        const float (*__restrict__ Bb)[LDSROW] =
            (const float (*)[LDSROW])Bbuf[kc & 1];
#pragma unroll
        for (int j = 0; j < KC / 4; ++j) {
            const int r = 4 * j + 2 * half;
#pragma unroll
            for (int tt = 0; tt < NT; ++tt) {
                v2f b;
                b.x = Bb[r][l16 + tt * 16];
                b.y = Bb[r + 1][l16 + tt * 16];
                acc[tt] = __builtin_amdgcn_wmma_f32_16x16x4_f32(
                    false, afrag[j], false, b, (short)0, acc[tt], false, false);
            }
        }
    }

#pragma unroll
    for (int tt = 0; tt < NT; ++tt) {
#pragma unroll
        for (int v = 0; v < 8; ++v) {
            const int row = m0 + v + 8 * half;
            const int col = n0 + tt * 16 + l16;
            C[(size_t)row * N + col] = acc[tt][v];
        }
    }
}

// ---------------------------------------------------------------------------
// Transpose per batch: VW [c=256][s=1024] -> VT [s=1024][c=256]
// so the sampler reads 256 contiguous channels per spatial corner.
// ---------------------------------------------------------------------------
__global__ void transpose_vw(const float* __restrict__ vw,
                             float* __restrict__ vt) {
    __shared__ float tile[32][33];
    const int b  = blockIdx.z;
    const int s0 = blockIdx.x * 32;      // 32 blocks over s
    const int c0 = blockIdx.y * 32;      // 8 blocks over c
    const float* __restrict__ src = vw + (size_t)b * VWSZ;
    float* __restrict__ dst       = vt + (size_t)b * VWSZ;
    const int tx = threadIdx.x;          // 32
    const int ty = threadIdx.y;          // 8
#pragma unroll
    for (int j = 0; j < 32; j += 8)
        tile[ty + j][tx] = src[(size_t)(c0 + ty + j) * 1024 + s0 + tx];
    __syncthreads();
#pragma unroll
    for (int j = 0; j < 32; j += 8)
        dst[(size_t)(s0 + ty + j) * 256 + c0 + tx] = tile[tx][ty + j];
}

// ---------------------------------------------------------------------------
// Softmax + bilinear grid-sample + weighted sum.
// One 256-thread block per (b, n): thread c owns output channel c.
// ---------------------------------------------------------------------------
struct SampleP { int x0, y0; float wx, wy; };

__global__ void sample_kernel(const float* __restrict__ qa,
                              const float* __restrict__ vt,
                              float* __restrict__ out) {
    const int bn = blockIdx.x;           // b*1024 + n
    const int b  = bn >> 10;
    const int n  = bn & 1023;
    const int c  = threadIdx.x;

    __shared__ float aw[NPTS];
    __shared__ SampleP spt[NPTS];

    const float* __restrict__ row = qa + (size_t)bn * 32;

    if (threadIdx.x == 0) {
        float l[NPTS];
        float m = -INFINITY;
#pragma unroll
        for (int k = 0; k < NPTS; ++k) { l[k] = row[k]; m = fmaxf(m, l[k]); }
        float s = 0.0f;
#pragma unroll
        for (int k = 0; k < NPTS; ++k) { l[k] = __expf(l[k] - m); s += l[k]; }
        float inv = 1.0f / s;
#pragma unroll
        for (int k = 0; k < NPTS; ++k) aw[k] = l[k] * inv;
    }
    if (threadIdx.x < NPTS) {
        const int k = threadIdx.x;
        // reference points: q = i*P + j -> ref = (lin[i], lin[j]), lin = 0..1
        const float refx = (float)(n >> 5) * (1.0f / 31.0f);
        const float refy = (float)(n & 31) * (1.0f / 31.0f);
        const float locx = refx + row[8 + 2 * k];
        const float locy = refy + row[8 + 2 * k + 1];
        const float gridx = (locx * (1.0f / 31.0f) - 0.5f) * 2.0f;
        const float gridy = (locy * (1.0f / 31.0f) - 0.5f) * 2.0f;
        const float gx = (gridx + 1.0f) * (PP * 0.5f) - 0.5f;
        const float gy = (gridy + 1.0f) * (PP * 0.5f) - 0.5f;
        const float x0f = floorf(gx);
        const float y0f = floorf(gy);
        spt[k].x0 = (int)x0f;
        spt[k].y0 = (int)y0f;
        spt[k].wx = gx - x0f;
        spt[k].wy = gy - y0f;
    }
    __syncthreads();

    const float* __restrict__ vb = vt + (size_t)b * VWSZ;
    float acc = 0.0f;
#pragma unroll
    for (int k = 0; k < NPTS; ++k) {
        const int   x0 = spt[k].x0, y0 = spt[k].y0;
        const float wx = spt[k].wx, wy = spt[k].wy;
        const float a  = aw[k];
        float v00 = 0.0f, v10 = 0.0f, v01 = 0.0f, v11 = 0.0f;
        const bool xin0 = (x0 >= 0)     && (x0 < PP);
        const bool xin1 = (x0 + 1 >= 0) && (x0 + 1 < PP);
        if ((y0 >= 0) && (y0 < PP)) {
            const float* r0 = vb + ((size_t)y0 * PP) * 256;
            if (xin0) v00 = r0[(size_t)x0 * 256 + c];
            if (xin1) v10 = r0[(size_t)(x0 + 1) * 256 + c];
        }
        if ((y0 + 1 >= 0) && (y0 + 1 < PP)) {
            const float* r1 = vb + ((size_t)(y0 + 1) * PP) * 256;
            if (xin0) v01 = r1[(size_t)x0 * 256 + c];
            if (xin1) v11 = r1[(size_t)(x0 + 1) * 256 + c];
        }
        const float sampled = v00 * (1.0f - wx) * (1.0f - wy)
                            + v10 * wx          * (1.0f - wy)
                            + v01 * (1.0f - wx) * wy
                            + v11 * wx          * wy;
        acc += a * sampled;
    }
    out[(size_t)bn * 256 + c] = acc;
}

// ---------------------------------------------------------------------------
// Launch
// ---------------------------------------------------------------------------
extern "C" void kernel_launch(void* const* d_in, const int* in_sizes, int n_in,
                              void* d_out, int out_size, void* d_ws, size_t ws_size,
                              hipStream_t stream) {
    const float* query    = (const float*)d_in[0];   // [8, 1024, 1024]
    const float* value    = (const float*)d_in[1];   // [8, 1024, 1024]
    const float* w_offset = (const float*)d_in[2];   // [1024, 16]
    const float* w_att    = (const float*)d_in[3];   // [1024, 8]
    const float* w_value  = (const float*)d_in[4];   // [1024, 256]
    float* out = (float*)d_out;                      // [8, 1024, 256]

    char* ws = (char*)d_ws;
    float* QA = (float*)ws;                                        // 8192*32   = 1 MB
    float* VW = (float*)(ws + (size_t)ROWS * 32 * 4);              // 8192*256  = 8 MB
    float* VT = (float*)(ws + ((size_t)ROWS * 32 + (size_t)BB * VWSZ) * 4);       // 8 MB
    float* W1 = (float*)(ws + ((size_t)ROWS * 32 + 2 * (size_t)BB * VWSZ) * 4);   // 128 KB

    // 1. pack projection weights [1024, 32]
    pack_w1<<<dim3((DIN * 32 + 255) / 256), dim3(256), 0, stream>>>(w_att, w_offset, W1);

    // 2. QA = query @ W1   (8192 x 1024 x 32), NT=2 -> grid.y = 1
    gemm_f32_wmma<2><<<dim3(ROWS / 128, 1), dim3(256), 0, stream>>>(
        query, W1, QA, ROWS, 32, DIN);

    // 3. VW = value @ w_value (8192 x 1024 x 256), NT=4 -> grid.y = 4
    gemm_f32_wmma<4><<<dim3(ROWS / 128, 4), dim3(256), 0, stream>>>(
        value, w_value, VW, ROWS, DOUT, DIN);

    // 4. per-batch transpose [c][s] -> [s][c]
    transpose_vw<<<dim3(32, 8, BB), dim3(32, 8), 0, stream>>>(VW, VT);

    // 5. softmax + bilinear sample + weighted sum
    sample_kernel<<<dim3(ROWS), dim3(256), 0, stream>>>(QA, VT, out);
}